// MorphoGNN_85933705659009
// MI455X (gfx1250) — compile-verified
//
#include <hip/hip_runtime.h>
#include <hip/hip_bf16.h>

#define D 128

typedef float v2f __attribute__((ext_vector_type(2)));
typedef float v4f __attribute__((ext_vector_type(4)));
typedef float v8f __attribute__((ext_vector_type(8)));

// ---------------- degree / normalization ----------------
__global__ void deg_init_kernel(float* deg, int n) {
    int i = blockIdx.x * blockDim.x + threadIdx.x;
    if (i < n) deg[i] = 1.0f;  // self loop contributes 1
}

__global__ void deg_count_kernel(const int* __restrict__ dst, float* deg, int E) {
    int e = blockIdx.x * blockDim.x + threadIdx.x;
    if (e < E) atomicAdd(&deg[dst[e]], 1.0f);
}

__global__ void deg_rsqrt_kernel(float* deg, int n) {
    int i = blockIdx.x * blockDim.x + threadIdx.x;
    if (i < n) deg[i] = rsqrtf(deg[i]);  // deg >= 1 always (self loops)
}

// ---------------- embedding gather ----------------
// one thread per float4 chunk: n*32 threads
__global__ void embed_kernel(const int* __restrict__ x, const float* __restrict__ emb,
                             float* __restrict__ h, int n) {
    int idx = blockIdx.x * blockDim.x + threadIdx.x;
    if (idx >= n * 32) return;
    int node = idx >> 5;
    int c = idx & 31;
    const v4f* src = (const v4f*)(emb + (size_t)x[node] * D) + c;
    v4f* dstp = (v4f*)(h + (size_t)node * D) + c;
    *dstp = *src;
}

// ---------------- WMMA GEMM: Out[n,128] = H[n,128] @ W[128,128] ----------------
// 256 threads = 8 waves; wave w computes rows [blk*128 + w*16, +16) x all 128 cols.
//
// W is staged into LDS pre-swizzled so that for k-group g (k = 4g..4g+3), lane l,
// tile t, the B-fragment pair {W[4g+koff][t*16+m], W[4g+koff+1][t*16+m]} with
// koff = (l>=16)?2:0, m = l&15 sits at a contiguous 8-byte-aligned LDS address:
//     lW[((g*32 + l)*8 + t)*2 + j]
// -> each B fragment is one ds_load_b64 into an even-aligned VGPR pair, and a
// lane's 8 tiles are 64 contiguous bytes (mergeable into ds_load_b128).
__launch_bounds__(256)
__global__ void gemm128_wmma_kernel(const float* __restrict__ H, const float* __restrict__ W,
                                    float* __restrict__ Out, int n) {
    __shared__ float lW[D * D];  // 64 KB swizzled copy of W
    const int tid = threadIdx.x;
    // cooperative swizzled fill: flat index i = (((g*32 + l)*8 + t)*2 + j)
    for (int i = tid; i < D * D; i += 256) {
        int j = i & 1;
        int t = (i >> 1) & 7;
        int l = (i >> 4) & 31;
        int g = i >> 9;
        int k = 4 * g + ((l >> 4) << 1) + j;
        lW[i] = W[k * D + t * 16 + (l & 15)];
    }
    __syncthreads();

    const int wave = tid >> 5;
    const int lane = tid & 31;
    const int rowBase = blockIdx.x * 128 + wave * 16;
    const int m = lane & 15;
    const int koff = (lane >> 4) << 1;  // lanes 0-15 -> K 0,1 ; lanes 16-31 -> K 2,3

    // clamp A row for out-of-range rows (EXEC must be all-1s for WMMA; stores predicated)
    int arow = rowBase + m;
    if (arow >= n) arow = n - 1;
    const float* __restrict__ Arow = H + (size_t)arow * D;

    v8f acc[8];
#pragma unroll
    for (int t = 0; t < 8; ++t) acc[t] = (v8f){0.f, 0.f, 0.f, 0.f, 0.f, 0.f, 0.f, 0.f};

    for (int g = 0; g < 32; ++g) {
        // A fragment (16x4 f32): lane holds {A[m][4g+koff], A[m][4g+koff+1]}
        v2f a = *(const v2f*)(Arow + 4 * g + koff);
        // B fragments: 8 contiguous aligned pairs in LDS for this (g, lane)
        const v2f* __restrict__ bp = (const v2f*)(lW + (g * 32 + lane) * 16);
#pragma unroll
        for (int t = 0; t < 8; ++t) {
            v2f b = bp[t];
            acc[t] = __builtin_amdgcn_wmma_f32_16x16x4_f32(
                /*neg_a=*/false, a, /*neg_b=*/false, b,
                /*c_mod=*/(short)0, acc[t], /*reuse_a=*/false, /*reuse_b=*/false);
        }
    }

    // C/D layout: VGPR v -> row v (lanes 0-15) / row v+8 (lanes 16-31); col = lane&15 within tile
    const int col0 = lane & 15;
    const int rowAdd = (lane >> 4) * 8;
#pragma unroll
    for (int t = 0; t < 8; ++t) {
#pragma unroll
        for (int v = 0; v < 8; ++v) {
            int r = rowBase + rowAdd + v;
            if (r < n) Out[(size_t)r * D + t * 16 + col0] = acc[t][v];
        }
    }
}

// ---------------- aggregation ----------------
// self-loop term: out[i,:] = hw[i,:] * dinv[i]^2   (also initializes out)
__global__ void self_init_kernel(const float* __restrict__ hw, const float* __restrict__ dinv,
                                 float* __restrict__ out, int n) {
    int idx = blockIdx.x * blockDim.x + threadIdx.x;
    if (idx >= n * 32) return;
    int node = idx >> 5;
    int c = idx & 31;
    float s = dinv[node];
    s = s * s;
    v4f v = *((const v4f*)(hw + (size_t)node * D) + c);
    v4f r = {v.x * s, v.y * s, v.z * s, v.w * s};
    *((v4f*)(out + (size_t)node * D) + c) = r;
}

// one wave per edge: lane handles float4 chunk, 4 atomic f32 adds
__global__ void edge_scatter_kernel(const int* __restrict__ src, const int* __restrict__ dst,
                                    const float* __restrict__ dinv, const float* __restrict__ hw,
                                    float* __restrict__ out, int E) {
    long long idx = (long long)blockIdx.x * blockDim.x + threadIdx.x;
    if (idx >= (long long)E * 32) return;
    int e = (int)(idx >> 5);
    int c = (int)(idx & 31);
    int s = src[e];
    int t = dst[e];
    float norm = dinv[s] * dinv[t];
    v4f v = *((const v4f*)(hw + (size_t)s * D) + c);
    float* o = out + (size_t)t * D + c * 4;
    atomicAdd(o + 0, v.x * norm);
    atomicAdd(o + 1, v.y * norm);
    atomicAdd(o + 2, v.z * norm);
    atomicAdd(o + 3, v.w * norm);
}

__global__ void bias_relu_kernel(float* __restrict__ out, const float* __restrict__ b, int n) {
    int idx = blockIdx.x * blockDim.x + threadIdx.x;
    if (idx >= n * 32) return;
    int node = idx >> 5;
    int c = idx & 31;
    v4f v = *((v4f*)(out + (size_t)node * D) + c);
    const v4f bv = *((const v4f*)b + c);
    v4f r = {fmaxf(v.x + bv.x, 0.f), fmaxf(v.y + bv.y, 0.f),
             fmaxf(v.z + bv.z, 0.f), fmaxf(v.w + bv.w, 0.f)};
    *((v4f*)(out + (size_t)node * D) + c) = r;
}

// ---------------- final projection: out[i] = dot(h[i,:], Wf) + bf ----------------
// one wave (32 lanes) per node; float4 partials + wave32 shuffle reduction
__global__ void final_dot_kernel(const float* __restrict__ h, const float* __restrict__ Wf,
                                 const float* __restrict__ bf, float* __restrict__ out, int n) {
    int tid = threadIdx.x;
    int wave = tid >> 5;
    int lane = tid & 31;
    int node = blockIdx.x * 8 + wave;
    if (node >= n) return;
    v4f a = *((const v4f*)(h + (size_t)node * D) + lane);
    v4f w = *((const v4f*)Wf + lane);
    float s = a.x * w.x + a.y * w.y + a.z * w.z + a.w * w.w;
#pragma unroll
    for (int off = 16; off > 0; off >>= 1) s += __shfl_xor(s, off, 32);
    if (lane == 0) out[node] = s + bf[0];
}

extern "C" void kernel_launch(void* const* d_in, const int* in_sizes, int n_in,
                              void* d_out, int out_size, void* d_ws, size_t ws_size,
                              hipStream_t stream) {
    const int* x       = (const int*)d_in[0];
    const int* eidx    = (const int*)d_in[1];
    const float* emb   = (const float*)d_in[2];
    const float* W1    = (const float*)d_in[3];
    const float* b1    = (const float*)d_in[4];
    const float* W2    = (const float*)d_in[5];
    const float* b2    = (const float*)d_in[6];
    const float* Wf    = (const float*)d_in[7];
    const float* bf    = (const float*)d_in[8];
    float* out         = (float*)d_out;

    const int n = in_sizes[0];          // 100000 nodes
    const int E = in_sizes[1] / 2;      // 800000 edges
    const int* srcI = eidx;
    const int* dstI = eidx + E;

    // workspace layout: h0 | t1 | t2 | dinv  (~154 MB)
    float* h0   = (float*)d_ws;
    float* t1   = h0 + (size_t)n * D;
    float* t2   = t1 + (size_t)n * D;
    float* dinv = t2 + (size_t)n * D;

    const int B = 256;
    dim3 blk(B);
    int gN   = (n + B - 1) / B;                         // per-node
    int gE   = (E + B - 1) / B;                         // per-edge
    int gN32 = (int)(((long long)n * 32 + B - 1) / B);  // per-float4-chunk
    long long tE32 = (long long)E * 32;
    int gE32 = (int)((tE32 + B - 1) / B);               // per-edge-chunk
    int gG   = (n + 127) / 128;                         // gemm row blocks
    int gF   = (n + 7) / 8;                             // final: 8 nodes/block

    // normalization
    deg_init_kernel<<<gN, blk, 0, stream>>>(dinv, n);
    deg_count_kernel<<<gE, blk, 0, stream>>>(dstI, dinv, E);
    deg_rsqrt_kernel<<<gN, blk, 0, stream>>>(dinv, n);

    // h0 = emb[x]
    embed_kernel<<<gN32, blk, 0, stream>>>(x, emb, h0, n);

    // layer 1: t1 = h0 @ W1 ; t2 = aggregate(t1) ; t2 = relu(t2 + b1)
    gemm128_wmma_kernel<<<gG, blk, 0, stream>>>(h0, W1, t1, n);
    self_init_kernel<<<gN32, blk, 0, stream>>>(t1, dinv, t2, n);
    edge_scatter_kernel<<<gE32, blk, 0, stream>>>(srcI, dstI, dinv, t1, t2, E);
    bias_relu_kernel<<<gN32, blk, 0, stream>>>(t2, b1, n);

    // layer 2: h0 = t2 @ W2 ; t1 = aggregate(h0) ; t1 = relu(t1 + b2)
    gemm128_wmma_kernel<<<gG, blk, 0, stream>>>(t2, W2, h0, n);
    self_init_kernel<<<gN32, blk, 0, stream>>>(h0, dinv, t1, n);
    edge_scatter_kernel<<<gE32, blk, 0, stream>>>(srcI, dstI, dinv, h0, t1, E);
    bias_relu_kernel<<<gN32, blk, 0, stream>>>(t1, b2, n);

    // final projection
    final_dot_kernel<<<gF, blk, 0, stream>>>(t1, Wf, bf, out, n);
}